// EventWarping_44822278701624
// MI455X (gfx1250) — compile-verified
//
#include <hip/hip_runtime.h>
#include <hip/hip_bf16.h>

#define IMG_HW 65536
#define IMG_W  256
#define EPSF   1e-9f

typedef __attribute__((ext_vector_type(2))) float v2f;
typedef __attribute__((ext_vector_type(8))) float v8f;

// Non-returning fp32 atomic add -> global_atomic_add_f32 (STOREcnt, no return).
__device__ __forceinline__ void atomic_add_f32(float* p, float v) {
    __hip_atomic_fetch_add(p, v, __ATOMIC_RELAXED, __HIP_MEMORY_SCOPE_AGENT);
}

// ---------------------------------------------------------------------------
// Phase 0: zero the 16 accumulator images (4 MB) and the scalar output.
// ---------------------------------------------------------------------------
__global__ void ew_zero(float* __restrict__ acc, float* __restrict__ out, int n) {
    int i = blockIdx.x * blockDim.x + threadIdx.x;
    if (i < n) acc[i] = 0.0f;
    if (i == 0) out[0] = 0.0f;
}

// ---------------------------------------------------------------------------
// Phase 1: fused fw+bw bilinear splat. One thread per event; 16 L2 atomics.
// acc layout: group g = (t*2 + b), t=0 -> tref=1, t=1 -> tref=0, each group = 4 images of 65536 floats:
//   [iwe_pos, iwe_neg, tsiwe_pos, tsiwe_neg]
// ---------------------------------------------------------------------------
__global__ __launch_bounds__(256) void ew_scatter(const float4* __restrict__ ev,
                                                  const float2* __restrict__ fl,
                                                  float* __restrict__ acc,
                                                  int total, int nper) {
    int gid = blockIdx.x * blockDim.x + threadIdx.x;
    if (gid >= total) return;
    float4 e = ev[gid];          // (ts, x, y, p)
    float2 f = fl[gid];          // (fx, fy)
    int b = gid / nper;          // batch index
    float ts = e.x, x = e.y, y = e.z;
    int pol = (e.w > 0.0f) ? 0 : 1;   // p is exactly +1 / -1

#pragma unroll
    for (int t = 0; t < 2; ++t) {      // t=0 -> tref=1 (fw), t=1 -> tref=0 (bw)
        float tref = (t == 0) ? 1.0f : 0.0f;
        float dt = tref - ts;
        float wx = fmaf(dt * 256.0f, f.x, x);   // FLOW_SCALING = 256
        float wy = fmaf(dt * 256.0f, f.y, y);
        float x0 = floorf(wx), y0 = floorf(wy);
        float dx = wx - x0,    dy = wy - y0;
        float tsw = 1.0f - fabsf(dt);           // max_ts = 1

        float* iweP = acc + (size_t)((t * 2 + b) * 4 + pol) * IMG_HW;
        float* tsP  = iweP + 2 * IMG_HW;

        float cw[4] = { (1.f - dx) * (1.f - dy), dx * (1.f - dy),
                        (1.f - dx) * dy,         dx * dy };
        float cx[4] = { x0, x0 + 1.f, x0, x0 + 1.f };
        float cy[4] = { y0, y0, y0 + 1.f, y0 + 1.f };
#pragma unroll
        for (int k = 0; k < 4; ++k) {
            if (cx[k] >= 0.f && cx[k] <= 255.f && cy[k] >= 0.f && cy[k] <= 255.f) {
                int idx = (int)cy[k] * IMG_W + (int)cx[k];
                atomic_add_f32(iweP + idx, cw[k]);
                atomic_add_f32(tsP  + idx, cw[k] * tsw);
            }
        }
    }
}

// ---------------------------------------------------------------------------
// Wave-level (32 lane) sum via V_WMMA_F32_16X16X4_F32 against all-ones B.
// A: lane L<16 holds A[L][0]=v (K=1,3 zeroed); lane L>=16 holds A[L-16][2]=v.
// With B == ones: D[m][n] = sum_k A[m][k] = v[m] + v[m+16] for every n.
// C/D layout: vgpr j, lanes 0-15 -> D[j][n]; lanes 16-31 -> D[8+j][n].
// So sum(c[0..7]) at lane0 + sum(c[0..7]) at lane16 = total over 32 lanes.
// Requires EXEC all-ones: callers have uniform control flow.
// ---------------------------------------------------------------------------
__device__ __forceinline__ float wave_sum32(float v) {
    v2f a;  a[0] = v;    a[1] = 0.0f;
    v2f bo; bo[0] = 1.0f; bo[1] = 1.0f;
    v8f c = {};
    c = __builtin_amdgcn_wmma_f32_16x16x4_f32(false, a, false, bo,
                                              (short)0, c, false, false);
    float s = c[0] + c[1] + c[2] + c[3] + c[4] + c[5] + c[6] + c[7];
    return __shfl(s, 0, 32) + __shfl(s, 16, 32);
}

// ---------------------------------------------------------------------------
// Phase 2: one workgroup per (tref, batch) group: compute
//   R = sum_px_pol (ts/(iwe+eps))^2 ;  Nnz = #px with iwe_pos+iwe_neg > 0
//   E_pol = sum_px exp(-0.6*iwe_pol)
//   term  = R/(Nnz+eps) + 65536/E0 + 65536/E1 - 2
// and atomically add 0.5*term into out (fw+bw average, batch sum).
// ---------------------------------------------------------------------------
__global__ __launch_bounds__(256) void ew_reduce(const float* __restrict__ acc,
                                                 float* __restrict__ out) {
    const float* iwe0 = acc + (size_t)blockIdx.x * 4 * IMG_HW;
    const float* iwe1 = iwe0 + IMG_HW;
    const float* tsw0 = iwe0 + 2 * IMG_HW;
    const float* tsw1 = iwe0 + 3 * IMG_HW;

    float sr2 = 0.f, snz = 0.f, se0 = 0.f, se1 = 0.f;
    for (int px = threadIdx.x; px < IMG_HW; px += 256) {
        __builtin_prefetch(&iwe0[px + 2048], 0, 0);   // global_prefetch_b8
        float ip = iwe0[px], iq = iwe1[px];
        float tp = tsw0[px], tq = tsw1[px];
        float rp = tp / (ip + EPSF);
        float rq = tq / (iq + EPSF);
        sr2 += rp * rp + rq * rq;
        snz += ((ip + iq) > 0.f) ? 1.f : 0.f;
        se0 += expf(-0.6f * ip);
        se1 += expf(-0.6f * iq);
    }

    // 32-lane reductions through the WMMA unit (EXEC all-ones here).
    sr2 = wave_sum32(sr2);
    snz = wave_sum32(snz);
    se0 = wave_sum32(se0);
    se1 = wave_sum32(se1);

    __shared__ float sm[4][8];
    int wave = threadIdx.x >> 5, lane = threadIdx.x & 31;
    if (lane == 0) {
        sm[0][wave] = sr2; sm[1][wave] = snz;
        sm[2][wave] = se0; sm[3][wave] = se1;
    }
    __syncthreads();
    if (threadIdx.x == 0) {
        float R = 0.f, Z = 0.f, E0 = 0.f, E1 = 0.f;
        for (int w = 0; w < 8; ++w) {
            R += sm[0][w]; Z += sm[1][w]; E0 += sm[2][w]; E1 += sm[3][w];
        }
        float term = R / (Z + EPSF) + 65536.0f / E0 + 65536.0f / E1 - 2.0f;
        atomicAdd(out, 0.5f * term);
    }
}

// ---------------------------------------------------------------------------
extern "C" void kernel_launch(void* const* d_in, const int* in_sizes, int n_in,
                              void* d_out, int out_size, void* d_ws, size_t ws_size,
                              hipStream_t stream) {
    const float4* events = (const float4*)d_in[0];
    const float2* flow   = (const float2*)d_in[1];
    const float4* ngev   = (const float4*)d_in[2];
    const float2* ngfl   = (const float2*)d_in[3];
    float* acc = (float*)d_ws;      // 16 * 65536 floats = 4 MB
    float* out = (float*)d_out;

    int totalE = in_sizes[0] / 4;   // B*N events
    int nE     = totalE / 2;        // per-batch N   (B = 2)
    int totalG = in_sizes[2] / 4;   // B*NG events
    int nG     = totalG / 2;

    int accN = 16 * IMG_HW;
    ew_zero<<<(accN + 255) / 256, 256, 0, stream>>>(acc, out, accN);
    ew_scatter<<<(totalE + 255) / 256, 256, 0, stream>>>(events, flow, acc, totalE, nE);
    ew_scatter<<<(totalG + 255) / 256, 256, 0, stream>>>(ngev, ngfl, acc, totalG, nG);
    ew_reduce<<<4, 256, 0, stream>>>(acc, out);
}